// KnowledgeModule_31696858644646
// MI455X (gfx1250) — compile-verified
//
#include <hip/hip_runtime.h>
#include <math.h>

// Sum-product network (log semiring), fan-in 4 per node.
// Memory-bound gather/segment-reduce: no matmul anywhere -> WMMA not applicable.
// Optimized for MI455X: b128 index loads, non-temporal streaming of index
// arrays, global_prefetch on the index stream, gather tables left RT so they
// stay resident in the 192MB L2.

typedef int v4i __attribute__((ext_vector_type(4)));

#define LOG2F_CONST 0.69314718055994530942f

// ---------------- encode: x = [-inf, 0, pos_i, log1mexp(pos_i), ...] --------
__global__ __launch_bounds__(256) void encode_kernel(
    const float* __restrict__ w, float* __restrict__ x, int n_vars) {
  int i = blockIdx.x * blockDim.x + threadIdx.x;
  if (i >= n_vars) return;
  float p = w[i];  // p < 0 (log-probability)
  // log(1 - exp(p)), numerically stable split at -log 2 (matches reference)
  float neg = (p > -LOG2F_CONST) ? logf(-expm1f(p)) : log1pf(-expf(p));
  float2 v; v.x = p; v.y = neg;
  // x base is 256B aligned; +2 floats keeps 8B alignment for float2 stores
  reinterpret_cast<float2*>(x + 2)[i] = v;
  if (i == 0) { x[0] = -INFINITY; x[1] = 0.0f; }
}

// ---------------- product layer: out[j] = sum of 4 gathered log-values ------
__global__ __launch_bounds__(256) void prod_kernel(
    const float* __restrict__ x, const v4i* __restrict__ ptrs,
    float* __restrict__ out, int n) {
  int j = blockIdx.x * blockDim.x + threadIdx.x;
  if (j >= n) return;
  // one lane per wave warms the index stream ahead (global_prefetch_b8;
  // speculative, OOB silently dropped)
  if ((threadIdx.x & 31u) == 0u)
    __builtin_prefetch((const void*)(ptrs + j + 4096), 0, 0);
  // read-once index stream: non-temporal b128 load
  v4i p = __builtin_nontemporal_load(ptrs + j);
  float a = x[p.x];
  float b = x[p.y];
  float c = x[p.z];
  float d = x[p.w];
  out[j] = (a + b) + (c + d);   // -inf propagates correctly
}

// ---------------- sum layer: segment log-sum-exp over 4 children ------------
__global__ __launch_bounds__(256) void lse_kernel(
    const float* __restrict__ x, const v4i* __restrict__ ptrs,
    float* __restrict__ out, int n) {
  int j = blockIdx.x * blockDim.x + threadIdx.x;
  if (j >= n) return;
  if ((threadIdx.x & 31u) == 0u)
    __builtin_prefetch((const void*)(ptrs + j + 4096), 0, 0);
  v4i p = __builtin_nontemporal_load(ptrs + j);
  float g0 = x[p.x];
  float g1 = x[p.y];
  float g2 = x[p.z];
  float g3 = x[p.w];
  float m = fmaxf(fmaxf(g0, g1), fmaxf(g2, g3));
  float r;
  if (m == -INFINITY) {
    // all children -inf; reference: z=nan->0, s=4+eps, log(s)+(-inf) = -inf
    r = -INFINITY;
  } else {
    // g_k may still be -inf: expf(-inf) = 0, matches reference exactly
    float s = expf(g0 - m) + expf(g1 - m) + expf(g2 - m) + expf(g3 - m);
    r = logf(s + 1e-15f) + m;
  }
  out[j] = r;
}

static inline size_t pad64(size_t n) { return (n + 63u) & ~(size_t)63u; }

extern "C" void kernel_launch(void* const* d_in, const int* in_sizes, int n_in,
                              void* d_out, int out_size, void* d_ws, size_t ws_size,
                              hipStream_t stream) {
  // input order: weights, ptrs0, csr0, ptrs1, csr1, ptrs2, csr2, ptrs3, csr3,
  //              n0, n1, n2, n3   (n* live on device; derive sizes host-side)
  const float* w     = (const float*)d_in[0];
  const int*   ptrs0 = (const int*)d_in[1];
  const int*   ptrs1 = (const int*)d_in[3];
  const int*   ptrs2 = (const int*)d_in[5];
  const int*   ptrs3 = (const int*)d_in[7];

  const int n_vars = in_sizes[0];
  const int n0 = in_sizes[1] / 4;   // fan-in (deg) = 4
  const int n1 = in_sizes[3] / 4;
  const int n2 = in_sizes[5] / 4;
  const int n3 = in_sizes[7] / 4;

  // scratch layout (floats), 256B-aligned sections
  float* ws = (float*)d_ws;
  size_t c0 = pad64((size_t)2 + 2 * (size_t)n_vars);
  float* x0 = ws;                    // encoded inputs: 2 + 2*n_vars
  float* x1 = x0 + c0;               // layer0 out: n0
  float* x2 = x1 + pad64((size_t)n0);// layer1 out: n1
  float* x3 = x2 + pad64((size_t)n1);// layer2 out: n2
  float* y  = (float*)d_out;         // layer3 out: n3
  (void)ws_size; (void)n_in; (void)out_size;

  const dim3 blk(256);
  encode_kernel<<<dim3((n_vars + 255) / 256), blk, 0, stream>>>(w, x0, n_vars);
  prod_kernel  <<<dim3((n0 + 255) / 256), blk, 0, stream>>>(x0, (const v4i*)ptrs0, x1, n0);
  lse_kernel   <<<dim3((n1 + 255) / 256), blk, 0, stream>>>(x1, (const v4i*)ptrs1, x2, n1);
  prod_kernel  <<<dim3((n2 + 255) / 256), blk, 0, stream>>>(x2, (const v4i*)ptrs2, x3, n2);
  lse_kernel   <<<dim3((n3 + 255) / 256), blk, 0, stream>>>(x3, (const v4i*)ptrs3, y,  n3);
}